// WordSmoothCriterion_5755256177154
// MI455X (gfx1250) — compile-verified
//
#include <hip/hip_runtime.h>
#include <hip/hip_bf16.h>

// Problem constants (from reference)
#define B_      4
#define T_      2048
#define V_      32000
#define NNZ_    801
#define NROWS_  (B_ * T_)          // 8192
#define TAU_    0.8f
#define RARE_   1.0f
#define ALPHA_  0.7f

#define WAVES_PER_BLOCK 4
#define BDIM (WAVES_PER_BLOCK * 32)

typedef __attribute__((ext_vector_type(4))) unsigned int v4u_t;
typedef __attribute__((ext_vector_type(8))) int          v8i_t;
typedef __attribute__((ext_vector_type(4))) int          v4i_t;

// Issue one TDM DMA: copy `nelem` contiguous 4-byte elements from global
// (gbase + row_byte_off) into LDS at byte offset lds_off.
// D# layout per CDNA5 ISA ch. 8 (CDNA5 ISA):
//   group0: [1:0]=count=1, [63:32]=lds_addr, [120:64]=global_addr, [127:126]=type=2
//   group1: [17:16]=data_size(2->4B), [79:48]=tensor_dim0, [127:112]=tile_dim0,
//           [207:160]=tensor_dim0_stride; everything else 0 (1-D tile, no pad,
//           no multicast, no iterate, no gather)
//   group2/3 (+extra group): zero (<=2D tensor)
__device__ __forceinline__ void tdm_load_row_b32(unsigned lds_off,
                                                 const void* gbase,
                                                 unsigned long long row_byte_off,
                                                 unsigned nelem) {
  unsigned long long ga = (unsigned long long)gbase + row_byte_off;
  v4u_t g0;
  g0.x = 1u;                                            // count = 1 (valid D#)
  g0.y = lds_off;                                       // lds_addr (bytes)
  g0.z = (unsigned)(ga & 0xFFFFFFFFull);                // global_addr[31:0]
  g0.w = (unsigned)((ga >> 32) & 0x01FFFFFFull)         // global_addr[56:32]
       | 0x80000000u;                                   // type = 2 ("image")
  v8i_t g1;
  g1[0] = 0x00020000;                  // data_size = 2 -> 4 bytes; mask/flags = 0
  g1[1] = (int)(nelem << 16);          // tensor_dim0[15:0] at bits[63:48]
  g1[2] = 0;                           // tensor_dim0 hi, tensor_dim1 lo
  g1[3] = (int)(nelem << 16);          // tile_dim0 at bits[127:112]
  g1[4] = 0;                           // tile_dim1 = 0 (unused), tile_dim2 = 0
  g1[5] = (int)nelem;                  // tensor_dim0_stride[31:0]
  g1[6] = 0;
  g1[7] = 0;
  v4i_t z4 = {0, 0, 0, 0};
  v8i_t z8 = {0, 0, 0, 0, 0, 0, 0, 0};
  // clang-23 / therock-10.0 form: 6 args (g0, g1, g2, g3, extra, cpol)
  __builtin_amdgcn_tensor_load_to_lds(g0, g1, z4, z4, z8, /*cpol=*/0);
}

// One wave (wave32) per token row n:
//   - 3 TDM DMAs stage sim_values[row], idf_values[row], sim_cols[row] into LDS
//   - lanes compute double-exp weights + gather logp[n, cols[j]]
//   - wave-shuffle reduce S = sum(w), G = sum(g*w); lane 0 emits G/S and the
//     masked ML term for this row. No cross-wave sync needed anywhere.
__global__ __launch_bounds__(BDIM) void wsc_row_kernel(
    const float* __restrict__ logp,        // [NROWS_, V_]
    const float* __restrict__ mask,        // [NROWS_]
    const float* __restrict__ sim_values,  // [V_, NNZ_]
    const float* __restrict__ idf_values,  // [V_, NNZ_]
    const int*   __restrict__ target,      // [NROWS_]
    const int*   __restrict__ sim_cols,    // [V_, NNZ_]
    float* __restrict__ rowG,              // [NROWS_] out: G/S per row
    float* __restrict__ rowML)             // [NROWS_] out: logp[n,row]*m per row
{
  __shared__ float lds_sim[WAVES_PER_BLOCK][NNZ_];
  __shared__ float lds_idf[WAVES_PER_BLOCK][NNZ_];
  __shared__ int   lds_col[WAVES_PER_BLOCK][NNZ_];

  const int wave = threadIdx.x >> 5;
  const int lane = threadIdx.x & 31;
  const int n = blockIdx.x * WAVES_PER_BLOCK + wave;
  if (n >= NROWS_) return;

  const int row = target[n];  // uniform across the wave
  const unsigned long long roff = (unsigned long long)row * (NNZ_ * 4ull);

  // Async DMA the three row slices into this wave's LDS slab.
  tdm_load_row_b32((unsigned)(size_t)&lds_sim[wave][0], sim_values, roff, NNZ_);
  tdm_load_row_b32((unsigned)(size_t)&lds_idf[wave][0], idf_values, roff, NNZ_);
  tdm_load_row_b32((unsigned)(size_t)&lds_col[wave][0], sim_cols,   roff, NNZ_);
  __builtin_amdgcn_s_wait_tensorcnt(0);   // TENSORcnt is per-wave: no barrier needed

  const float inv_tau = 1.0f / TAU_;
  const float* __restrict__ lpn = logp + (size_t)n * V_;

  float S = 0.0f, G = 0.0f;
  for (int j = lane; j < NNZ_; j += 32) {
    float v = lds_sim[wave][j] - 1.0f - (TAU_ * RARE_) * lds_idf[wave][j];
    // faithful to reference: exp applied twice
    v = __expf(__expf(v * inv_tau) * inv_tau);
    S += v;
    const int c = lds_col[wave][j];
    G += lpn[c] * v;                      // random gather within the 128KB logp row
  }

  // wave32 shuffle reduction
  #pragma unroll
  for (int off = 16; off > 0; off >>= 1) {
    S += __shfl_down(S, off, 32);
    G += __shfl_down(G, off, 32);
  }

  if (lane == 0) {
    rowG[n]  = G / S;                     // == sum(g * sim) for this row
    rowML[n] = lpn[row] * mask[n];
  }
}

// Deterministic final reduction: one block sums the 8192 per-row partials and
// the mask (denom), then writes the scalar loss.
__global__ __launch_bounds__(256) void wsc_reduce_kernel(
    const float* __restrict__ rowG,
    const float* __restrict__ rowML,
    const float* __restrict__ mask,
    float* __restrict__ out)
{
  __shared__ float sG[8], sM[8], sD[8];
  float g = 0.0f, m = 0.0f, d = 0.0f;
  for (int i = threadIdx.x; i < NROWS_; i += 256) {
    g += rowG[i];
    m += rowML[i];
    d += mask[i];
  }
  #pragma unroll
  for (int off = 16; off > 0; off >>= 1) {
    g += __shfl_down(g, off, 32);
    m += __shfl_down(m, off, 32);
    d += __shfl_down(d, off, 32);
  }
  const int wave = threadIdx.x >> 5;
  const int lane = threadIdx.x & 31;
  if (lane == 0) { sG[wave] = g; sM[wave] = m; sD[wave] = d; }
  __syncthreads();
  if (threadIdx.x == 0) {
    float tg = 0.0f, tm = 0.0f, td = 0.0f;
    #pragma unroll
    for (int w = 0; w < 8; ++w) { tg += sG[w]; tm += sM[w]; td += sD[w]; }
    // loss = ALPHA * (-sum(g*sim)/denom) + (1-ALPHA) * (-sum(lp_t*m)/denom)
    out[0] = -(ALPHA_ * tg + (1.0f - ALPHA_) * tm) / td;
  }
}

extern "C" void kernel_launch(void* const* d_in, const int* in_sizes, int n_in,
                              void* d_out, int out_size, void* d_ws, size_t ws_size,
                              hipStream_t stream) {
  const float* logp       = (const float*)d_in[0];  // [B,T,V] f32
  const float* mask       = (const float*)d_in[1];  // [B,T]   f32
  const float* sim_values = (const float*)d_in[2];  // [V,NNZ] f32
  const float* idf_values = (const float*)d_in[3];  // [V,NNZ] f32
  const int*   target     = (const int*)d_in[4];    // [B,T]   i32
  const int*   sim_cols   = (const int*)d_in[5];    // [V,NNZ] i32
  float* out = (float*)d_out;

  // Workspace: 2 * 8192 floats = 64 KB of per-row partials (fully rewritten
  // every call -> no zero-init needed, deterministic across replays).
  float* rowG  = (float*)d_ws;
  float* rowML = rowG + NROWS_;

  wsc_row_kernel<<<dim3(NROWS_ / WAVES_PER_BLOCK), dim3(BDIM), 0, stream>>>(
      logp, mask, sim_values, idf_values, target, sim_cols, rowG, rowML);
  wsc_reduce_kernel<<<dim3(1), dim3(256), 0, stream>>>(rowG, rowML, mask, out);
}